// EMAVectorQuantizer_26938034881056
// MI455X (gfx1250) — compile-verified
//
#include <hip/hip_runtime.h>
#include <hip/hip_bf16.h>

typedef __attribute__((ext_vector_type(2))) float v2f;
typedef __attribute__((ext_vector_type(8))) float v8f;

#define NVEC   32768          // 8*4096 z vectors
#define DIM    256
#define KCODES 8192
#define NTILES (KCODES / 16)  // 512
#define DECAY_F 0.99f
#define ONE_M_DECAY 0.01f
#define EPS_F 1e-5f
#define LOSS_SCALE (0.25f / (float)(NVEC * DIM))   // COMMITMENT_COST / (N*D)

// ---- output layout (flat f32 concat, reference return order) ----
#define OFF_ZQ   ((size_t)0)
#define OFF_IDX  ((size_t)NVEC * DIM)                    // 8388608
#define OFF_LOSS (OFF_IDX + NVEC)                        // 8421376
#define OFF_EMB  (OFF_LOSS + 1)                          // 8421377
#define OFF_CS   (OFF_EMB + (size_t)KCODES * DIM)        // 10518529
#define OFF_EEMB (OFF_CS + KCODES)                       // 10526721

// ---- workspace layout (float units) ----
#define WS_DW   ((size_t)0)                  // KCODES*DIM
#define WS_E2   ((size_t)KCODES * DIM)       // KCODES
#define WS_CS   (WS_E2 + KCODES)             // KCODES
#define WS_N    (WS_CS + KCODES)             // 1 (+pad)
#define WS_IDX  (WS_N + 8)                   // NVEC ints

#define LDS_STRIDE 260                       // 260 % 64 == 4 -> conflict-free b64 reads

// CDNA5 async global->LDS copy (ASYNCcnt-tracked, no VGPR staging).
__device__ __forceinline__ void async_copy_b128(unsigned lds_byte_addr,
                                                const void* gaddr) {
  asm volatile("global_load_async_to_lds_b128 %0, %1, off"
               :: "v"(lds_byte_addr), "v"(gaddr)
               : "memory");
}

__device__ __forceinline__ void wait_asynccnt_le4() {
#if __has_builtin(__builtin_amdgcn_s_wait_asynccnt)
  __builtin_amdgcn_s_wait_asynccnt(4);
#else
  asm volatile("s_wait_asynccnt 4" ::: "memory");
#endif
}

// ---------------- e2[k] = sum(emb[k]^2) ----------------
__global__ __launch_bounds__(256) void e2_kernel(const float* __restrict__ emb,
                                                 float* __restrict__ e2) {
  __shared__ float red[8];
  int row = blockIdx.x;
  float v = emb[(size_t)row * DIM + threadIdx.x];
  float s = v * v;
  #pragma unroll
  for (int m = 16; m >= 1; m >>= 1) s += __shfl_xor(s, m, 32);
  if ((threadIdx.x & 31) == 0) red[threadIdx.x >> 5] = s;
  __syncthreads();
  if (threadIdx.x == 0) {
    float t = 0.f;
    #pragma unroll
    for (int i = 0; i < 8; ++i) t += red[i];
    e2[row] = t;
  }
}

// ---------------- main: WMMA distances + fused argmin ----------------
// Block = 256 threads = 8 waves; each wave owns 16 z-rows (A cached in 128
// VGPRs), block covers 128 rows. 512 code tiles of 16 streamed through an
// LDS double buffer filled by GLOBAL_LOAD_ASYNC_TO_LDS_B128 (tile t+1 in
// flight while tile t's 64 WMMAs run). 2 independent WMMA accumulator chains.
__global__ __launch_bounds__(256) void vq_argmin_wmma(
    const float* __restrict__ z, const float* __restrict__ emb,
    const float* __restrict__ e2, int* __restrict__ idx_i,
    float* __restrict__ idx_f) {
  __shared__ float ldsB[2][16 * LDS_STRIDE];

  const int lane = threadIdx.x & 31;
  const int wave = threadIdx.x >> 5;
  const int lmod = lane & 15;
  const int half = lane >> 4;                       // 0: K+0/1, 1: K+2/3
  const int m0 = (blockIdx.x * 8 + wave) * 16;      // first z-row of this wave

  // Cache A (16 rows x 256): per-lane 64 float2 (ISA 16x4 f32 A layout)
  v2f A[64];
  const float* zrow = z + (size_t)(m0 + lmod) * DIM + half * 2;
  #pragma unroll
  for (int s = 0; s < 64; ++s) A[s] = *(const v2f*)(zrow + s * 4);

  // async-staging map: thread -> 16 consecutive floats of one emb row
  const int sr = threadIdx.x >> 4;        // 0..15 (row in tile)
  const int sc = (threadIdx.x & 15) * 16; // 0..240 (col start)
  const unsigned ldsBase =
      (unsigned)(uintptr_t)&ldsB[0][0];                 // LDS byte offset
  const unsigned rowOff = (unsigned)((sr * LDS_STRIDE + sc) * 4);
  const unsigned bufStride = (unsigned)(16 * LDS_STRIDE * 4);

  // prologue: tile 0 -> buffer 0 (4x b128 per thread, ASYNCcnt += 4)
  {
    const float* g = emb + (size_t)sr * DIM + sc;
    #pragma unroll
    for (int q = 0; q < 4; ++q)
      async_copy_b128(ldsBase + rowOff + 16u * q, g + 4 * q);
  }

  float bestd[8];
  int   besti[8];
  #pragma unroll
  for (int m = 0; m < 8; ++m) { bestd[m] = 3.4e38f; besti[m] = 0; }

  for (int t = 0; t < NTILES; ++t) {
    // issue tile t+1 into the other buffer (wraps to tile 0 on the last
    // iteration so the ASYNCcnt bookkeeping stays uniform; dead buffer).
    {
      int tn = (t + 1 < NTILES) ? (t + 1) : 0;
      const float* g = emb + (size_t)(tn * 16 + sr) * DIM + sc;
      unsigned base = ldsBase + ((unsigned)((t + 1) & 1)) * bufStride + rowOff;
      #pragma unroll
      for (int q = 0; q < 4; ++q)
        async_copy_b128(base + 16u * q, g + 4 * q);
    }
    // async loads complete in order: <=4 outstanding means tile t landed.
    wait_asynccnt_le4();
    __syncthreads();   // all waves' tile-t data visible in LDS

    v8f acc0 = {0.f, 0.f, 0.f, 0.f, 0.f, 0.f, 0.f, 0.f};
    v8f acc1 = {0.f, 0.f, 0.f, 0.f, 0.f, 0.f, 0.f, 0.f};
    const float* brow = &ldsB[t & 1][lmod * LDS_STRIDE + half * 2];
    #pragma unroll
    for (int s = 0; s < 64; s += 2) {
      v2f b0 = *(const v2f*)(brow + s * 4);
      v2f b1 = *(const v2f*)(brow + s * 4 + 4);
      acc0 = __builtin_amdgcn_wmma_f32_16x16x4_f32(
          false, A[s], false, b0, (short)0, acc0, false, false);
      acc1 = __builtin_amdgcn_wmma_f32_16x16x4_f32(
          false, A[s + 1], false, b1, (short)0, acc1, false, false);
    }

    // dist(row, col) = e2[col] - 2*dot (row-constant |z|^2 dropped for argmin)
    float e2v = e2[t * 16 + lmod];
    int cand  = t * 16 + lmod;
    #pragma unroll
    for (int m = 0; m < 8; ++m) {
      float dot = acc0[m] + acc1[m];
      float d = fmaf(-2.0f, dot, e2v);
      if (d < bestd[m]) { bestd[m] = d; besti[m] = cand; }
    }
    __syncthreads();   // done reading buf[t&1] before iter t+1 overwrites it
  }

  // cross-lane reduce within each 16-lane half (C layout: lanes0-15 hold M=m,
  // lanes16-31 hold M=m+8); tie-break to lowest index like jnp.argmin.
  #pragma unroll
  for (int m = 0; m < 8; ++m) {
    float bd = bestd[m]; int bi = besti[m];
    #pragma unroll
    for (int mask = 1; mask <= 8; mask <<= 1) {
      float od = __shfl_xor(bd, mask, 32);
      int   oi = __shfl_xor(bi, mask, 32);
      if (od < bd || (od == bd && oi < bi)) { bd = od; bi = oi; }
    }
    bestd[m] = bd; besti[m] = bi;
  }
  if (lmod == 0) {
    int rowBase = m0 + half * 8;
    #pragma unroll
    for (int m = 0; m < 8; ++m) {
      idx_i[rowBase + m] = besti[m];
      idx_f[rowBase + m] = (float)besti[m];
    }
  }
}

// ---------------- scatter: cluster_size + dw (segment_sum) ----------------
__global__ __launch_bounds__(256) void scatter_kernel(
    const float* __restrict__ z, const int* __restrict__ idx,
    float* __restrict__ cs, float* __restrict__ dw) {
  size_t t = (size_t)blockIdx.x * 256 + threadIdx.x;   // over NVEC*DIM
  int n = (int)(t >> 8), d = (int)(t & 255);
  int k = idx[n];
  atomicAdd(&dw[(size_t)k * DIM + d], z[t]);
  if (d == 0) atomicAdd(&cs[k], 1.0f);
}

// ---------------- EMA updates ----------------
__global__ __launch_bounds__(256) void ema_kernel(
    const float* __restrict__ ema_cs, const float* __restrict__ cs,
    const float* __restrict__ ema_emb, const float* __restrict__ dw,
    float* __restrict__ out_necs, float* __restrict__ out_neemb) {
  size_t t = (size_t)blockIdx.x * 256 + threadIdx.x;   // over KCODES*DIM
  out_neemb[t] = ema_emb[t] * DECAY_F + dw[t] * ONE_M_DECAY;
  if (t < KCODES) out_necs[t] = ema_cs[t] * DECAY_F + cs[t] * ONE_M_DECAY;
}

// ---------------- n = sum(new_ema_cluster_size), single block ----------------
__global__ __launch_bounds__(256) void sum_n_kernel(const float* __restrict__ necs,
                                                    float* __restrict__ n_out) {
  __shared__ float red[8];
  float s = 0.f;
  for (int i = threadIdx.x; i < KCODES; i += 256) s += necs[i];
  #pragma unroll
  for (int m = 16; m >= 1; m >>= 1) s += __shfl_xor(s, m, 32);
  if ((threadIdx.x & 31) == 0) red[threadIdx.x >> 5] = s;
  __syncthreads();
  if (threadIdx.x == 0) {
    float t = 0.f;
    #pragma unroll
    for (int i = 0; i < 8; ++i) t += red[i];
    *n_out = t;
  }
}

// ---------------- new_embedding = new_ema_embedding / smoothed ----------------
__global__ __launch_bounds__(256) void newemb_kernel(
    const float* __restrict__ necs, const float* __restrict__ neemb,
    const float* __restrict__ n_ptr, float* __restrict__ out_emb) {
  size_t t = (size_t)blockIdx.x * 256 + threadIdx.x;   // over KCODES*DIM
  int k = (int)(t >> 8);
  float n = *n_ptr;
  float smoothed = (necs[k] + EPS_F) / (n + (float)KCODES * EPS_F) * n;
  out_emb[t] = neemb[t] / smoothed;
}

// ---------------- z_q_st (== z) copy + commitment loss ----------------
__global__ __launch_bounds__(256) void copyloss_kernel(
    const float* __restrict__ z, const float* __restrict__ emb,
    const int* __restrict__ idx, float* __restrict__ out_zq,
    float* __restrict__ out_loss) {
  __shared__ float red[8];
  size_t t = (size_t)blockIdx.x * 256 + threadIdx.x;   // over NVEC*DIM
  float zv = z[t];
  out_zq[t] = zv;
  int n = (int)(t >> 8), d = (int)(t & 255);
  float q = emb[(size_t)idx[n] * DIM + d];
  float diff = q - zv;
  float s = diff * diff;
  #pragma unroll
  for (int m = 16; m >= 1; m >>= 1) s += __shfl_xor(s, m, 32);
  if ((threadIdx.x & 31) == 0) red[threadIdx.x >> 5] = s;
  __syncthreads();
  if (threadIdx.x == 0) {
    float tt = 0.f;
    #pragma unroll
    for (int i = 0; i < 8; ++i) tt += red[i];
    atomicAdd(out_loss, tt * LOSS_SCALE);
  }
}

extern "C" void kernel_launch(void* const* d_in, const int* in_sizes, int n_in,
                              void* d_out, int out_size, void* d_ws, size_t ws_size,
                              hipStream_t stream) {
  const float* z       = (const float*)d_in[0];
  const float* emb     = (const float*)d_in[1];
  const float* ema_cs  = (const float*)d_in[2];
  const float* ema_emb = (const float*)d_in[3];

  float* out = (float*)d_out;
  float* ws  = (float*)d_ws;

  float* ws_dw = ws + WS_DW;
  float* ws_e2 = ws + WS_E2;
  float* ws_cs = ws + WS_CS;
  float* ws_n  = ws + WS_N;
  int*   ws_ix = (int*)(ws + WS_IDX);

  // zero accumulators (graph-capturable memset nodes)
  hipMemsetAsync(ws_dw, 0, (size_t)KCODES * DIM * sizeof(float), stream);
  hipMemsetAsync(ws_cs, 0, (size_t)KCODES * sizeof(float), stream);
  hipMemsetAsync(out + OFF_LOSS, 0, sizeof(float), stream);

  e2_kernel<<<KCODES, 256, 0, stream>>>(emb, ws_e2);

  vq_argmin_wmma<<<NVEC / 128, 256, 0, stream>>>(z, emb, ws_e2, ws_ix,
                                                 out + OFF_IDX);

  scatter_kernel<<<(NVEC * DIM) / 256, 256, 0, stream>>>(z, ws_ix, ws_cs, ws_dw);

  ema_kernel<<<(KCODES * DIM) / 256, 256, 0, stream>>>(
      ema_cs, ws_cs, ema_emb, ws_dw, out + OFF_CS, out + OFF_EEMB);

  sum_n_kernel<<<1, 256, 0, stream>>>(out + OFF_CS, ws_n);

  newemb_kernel<<<(KCODES * DIM) / 256, 256, 0, stream>>>(
      out + OFF_CS, out + OFF_EEMB, ws_n, out + OFF_EMB);

  copyloss_kernel<<<(NVEC * DIM) / 256, 256, 0, stream>>>(
      z, emb, ws_ix, out + OFF_ZQ, out + OFF_LOSS);
}